// CubeSimulator_6777458393800
// MI455X (gfx1250) — compile-verified
//
#include <hip/hip_runtime.h>
#include <math.h>

// ---------------------------------------------------------------------------
// Galaxy cube simulator, restructured as gridding (velocity histogram with
// linear binning) + Gaussian-kernel GEMM on v_wmma_f32_16x16x4_f32.
// ---------------------------------------------------------------------------

typedef float v2f __attribute__((ext_vector_type(2)));
typedef float v8f __attribute__((ext_vector_type(8)));

#define N_HI   192            // IMAGE_RES_OUT * IMAGE_UPSCALE
#define F_HI   128            // FREQ_RES_OUT * FREQ_UPSCALE
#define B_BINS 512            // velocity bins (K dimension of GEMM)
#define HSTR   (B_BINS + 4)   // LDS row stride in dwords (bank-conflict pad)
#define RES_R  1000           // RADIUS_RES
#define C_KMS  299792.458f
#define F_REST 230.538f

struct Params {
  float cr, sr, ci, si;       // rotation terms
  float L0, inv_dlog;         // log10(Rmin), 1/dlog for interp index
  float Vb, dv, inv_dv;       // bin range / width
  float sig_sq, norm;         // line broadening
  float invRd, invhz;         // intensity params
};

__device__ __forceinline__ float axisf(int i) {
  // mirrors make_spatial_axis: x_i = (-500 + dx) + dx*(i-1), dx = 1000/191
  const float DXF = 1000.0f / 191.0f;
  return (-500.0f + DXF) + DXF * ((float)i - 1.0f);
}

// ---------------------------------------------------------------------------
// Kernel 1: scalar setup + rotation-curve tables (single block)
// ---------------------------------------------------------------------------
__global__ __launch_bounds__(256) void setup_kernel(
    const float* inclination, const float* sky_rot, const float* line_broadening,
    const float* velocity_shift, const float* v_max, const float* r_c,
    const float* Rd_p, const float* hz_p, const float* freqs,
    Params* prm, float* vlab, float* Rgrid, float* vgrid)
{
  __shared__ Params sp;
  __shared__ float sh_misc[4];   // f0, f31, vshift, dlog
  const int t = threadIdx.x;
  if (t == 0) {
    const float inc = inclination[0], rot = sky_rot[0];
    const float cr = cosf(rot), sr = sinf(rot);
    const float ci = cosf(inc), si = sinf(inc);
    const float Rmin = 0.1f * (1000.0f / 192.0f);
    // R^2 = rot_x^2 + rot_y^2 is convex in (x,y,z) -> max over grid at a corner
    const float a0 = axisf(0), a1 = axisf(191);
    const float ex[2] = {a0, a1};
    float maxR2 = 0.0f;
    for (int i = 0; i < 2; ++i)
      for (int j = 0; j < 2; ++j) {
        const float x = ex[i], y = ex[j];
        const float rx = cr * x - sr * y;
        const float y1 = sr * x + cr * y;
        for (int k = 0; k < 2; ++k) {
          const float ry = ci * y1 - si * ex[k];
          maxR2 = fmaxf(maxR2, rx * rx + ry * ry);
        }
      }
    const float Rmax = sqrtf(maxR2);
    const float L0 = log10f(Rmin), L1 = log10f(Rmax);
    const float dlog = (L1 - L0) / 999.0f;
    const float vmax = v_max[0];
    const float sig  = line_broadening[0];
    const float sig_sq = sig * sig;
    const float Vb = vmax * 1.0001f;                 // |v_los| <= v_max strictly
    const float dv = 2.0f * Vb / (float)(B_BINS - 1);
    sp.cr = cr; sp.sr = sr; sp.ci = ci; sp.si = si;
    sp.L0 = L0; sp.inv_dlog = 1.0f / dlog;
    sp.Vb = Vb; sp.dv = dv; sp.inv_dv = 1.0f / dv;
    sp.sig_sq = sig_sq;
    sp.norm = 1.0f / sqrtf(2.0f * 3.14159265358979323846f * sig_sq);
    sp.invRd = 1.0f / Rd_p[0];
    sp.invhz = 1.0f / hz_p[0];
    sh_misc[0] = freqs[0];
    sh_misc[1] = freqs[31];
    sh_misc[2] = velocity_shift[0];
    sh_misc[3] = dlog;
    *prm = sp;
  }
  __syncthreads();
  const float L0 = sp.L0, dlog = sh_misc[3];
  const float vmax = v_max[0], rc = r_c[0];
  const float Rmin = 0.1f * (1000.0f / 192.0f);
  const float LN10 = 2.302585092994046f;
  for (int k = t; k < RES_R; k += 256) {
    const float R = expf(LN10 * (L0 + dlog * (float)k));   // 10^(...)
    Rgrid[k] = R;
    vgrid[k] = vmax * R / sqrtf(R * R + rc * rc + Rmin * Rmin);
  }
  if (t < F_HI) {
    const float f0 = sh_misc[0], f31 = sh_misc[1], vshift = sh_misc[2];
    const float step = (f31 - f0) / 127.0f;
    const float fu = f0 + step * (float)t;
    vlab[t] = C_KMS * (F_REST - fu) / F_REST - vshift;
  }
}

// ---------------------------------------------------------------------------
// Kernel 2: Gaussian weight matrix G[f,b] (norm folded in), 128 x 512
// ---------------------------------------------------------------------------
__global__ __launch_bounds__(512) void g_kernel(const Params* prm,
                                                const float* vlab, float* G)
{
  const int f = blockIdx.x;
  const int b = threadIdx.x;
  const float vb = -prm->Vb + prm->dv * (float)b;
  const float d  = vlab[f] - vb;
  G[f * B_BINS + b] = prm->norm * expf(-0.5f * d * d / prm->sig_sq);
}

// ---------------------------------------------------------------------------
// Kernel 3: fused histogram build + WMMA GEMM + 4x2x2 down-averaging.
// One WG (8 waves) = 4x4 pixel tile x all 128 frequencies.
// ---------------------------------------------------------------------------
__global__ __launch_bounds__(256) void cube_kernel(
    const Params* __restrict__ prm,
    const float* __restrict__ Rg_g, const float* __restrict__ vg_g,
    const float* __restrict__ G,    float* __restrict__ out)
{
  __shared__ float Hs[16 * HSTR];   // per-pixel velocity histograms
  __shared__ float Rg[RES_R];
  __shared__ float vg[RES_R];
  __shared__ float obuf[128];       // 32 fo x 2 x 2 output accumulators

  const int t = threadIdx.x;
  // uniform params
  const float cr = prm->cr, sr = prm->sr, ci = prm->ci, si = prm->si;
  const float L0 = prm->L0, inv_dlog = prm->inv_dlog;
  const float Vb = prm->Vb, inv_dv = prm->inv_dv;
  const float invRd = prm->invRd, invhz = prm->invhz;

  for (int i = t; i < 16 * HSTR; i += 256) Hs[i] = 0.0f;
  for (int i = t; i < RES_R; i += 256) { Rg[i] = Rg_g[i]; vg[i] = vg_g[i]; }
  if (t < 128) obuf[t] = 0.0f;
  __syncthreads();

  // ---------------- Phase 1: build 16 x 512 histogram --------------------
  const int p  = t & 15;
  const int px = p & 3, py = p >> 2;
  const int ix = blockIdx.x * 4 + px;
  const int iy = blockIdx.y * 4 + py;
  const float xc = axisf(ix), yc = axisf(iy);
  const float y1 = sr * xc + cr * yc;     // z-independent
  const float rx = cr * xc - sr * yc;     // z-independent (rot_x)
  const float Rg0 = Rg[0], Rg999 = Rg[RES_R - 1];
  const float vg0 = vg[0], vg999 = vg[RES_R - 1];

  for (int iz = (t >> 4); iz < N_HI; iz += 16) {
    const float zc = axisf(iz);
    const float ry = ci * y1 - si * zc;   // rot_y
    const float rz = si * y1 + ci * zc;   // rot_z
    const float R  = sqrtf(rx * rx + ry * ry);
    // jnp.interp(R, R_grid, v_grid) on exactly log-spaced grid
    float v;
    if (R <= Rg0)        v = vg0;
    else if (R >= Rg999) v = vg999;
    else {
      float tt = (log10f(R) - L0) * inv_dlog;
      int k = (int)tt;
      k = max(0, min(k, RES_R - 2));
      if (R < Rg[k] && k > 0) --k;
      else if (R > Rg[k + 1] && k < RES_R - 2) ++k;
      const float Rk = Rg[k], Rk1 = Rg[k + 1];
      const float w = (R - Rk) / (Rk1 - Rk);
      v = vg[k] + (vg[k + 1] - vg[k]) * w;
    }
    // v_los = -si * v_abs * cos(theta), cos(atan2(ry,rx)) = rx/R
    const float invR = (R > 0.0f) ? (1.0f / R) : 0.0f;
    const float vlos = -si * v * rx * invR;
    const float zn = rz * invhz;
    const float inten = expf(-R * invRd - 0.5f * zn * zn);
    // linear binning (2nd-order accurate deposition)
    const float tb = (vlos + Vb) * inv_dv;
    int b0 = (int)floorf(tb);
    b0 = max(0, min(b0, B_BINS - 2));
    float w1 = tb - (float)b0;
    w1 = fminf(fmaxf(w1, 0.0f), 1.0f);
    atomicAdd(&Hs[p * HSTR + b0],     inten * (1.0f - w1));
    atomicAdd(&Hs[p * HSTR + b0 + 1], inten * w1);
  }
  __syncthreads();

  // ---------------- Phase 2: GEMM cube[f,p] = sum_b G[f,b] * H[p,b] ------
  // wave w -> freq tile [16w,16w+16); lanes: m = lane&15, hs = lane>>4
  const int lane = t & 31;
  const int w    = t >> 5;
  const int m    = lane & 15;
  const int hs   = lane >> 4;
  v8f acc = {};
  const float* Arow = G + (size_t)(w * 16 + m) * B_BINS + 2 * hs;  // A: G[f][k]
  const float* Brow = &Hs[m * HSTR + 2 * hs];                       // B: H[p][k]
  #pragma unroll 4
  for (int k = 0; k < B_BINS; k += 4) {
    const v2f a = *(const v2f*)(Arow + k);   // lane m: K = 2*hs, 2*hs+1
    const v2f b = *(const v2f*)(Brow + k);
    acc = __builtin_amdgcn_wmma_f32_16x16x4_f32(false, a, false, b,
                                                (short)0, acc, false, false);
  }

  // ---------------- Phase 3: 4(freq) x 2 x 2 (spatial) averaging ---------
  // D layout: acc[r] = cube[f = 16w + r + 8*hs][pixel n = lane&15]
  const int xo = px >> 1, yo = py >> 1;
  #pragma unroll
  for (int r = 0; r < 8; ++r) {
    const int f  = w * 16 + r + 8 * hs;
    const int fo = f >> 2;
    atomicAdd(&obuf[fo * 4 + xo * 2 + yo], acc[r] * (1.0f / 16.0f));
  }
  __syncthreads();
  if (t < 128) {
    const int fo  = t >> 2;
    const int xo2 = (t >> 1) & 1;
    const int yo2 = t & 1;
    const int OX = blockIdx.x * 2 + xo2;
    const int OY = blockIdx.y * 2 + yo2;
    out[fo * 96 * 96 + OX * 96 + OY] = obuf[t];
  }
}

// ---------------------------------------------------------------------------
extern "C" void kernel_launch(void* const* d_in, const int* in_sizes, int n_in,
                              void* d_out, int out_size, void* d_ws, size_t ws_size,
                              hipStream_t stream)
{
  (void)in_sizes; (void)n_in; (void)out_size; (void)ws_size;
  const float* inclination     = (const float*)d_in[0];
  const float* sky_rot         = (const float*)d_in[1];
  const float* line_broadening = (const float*)d_in[2];
  const float* velocity_shift  = (const float*)d_in[3];
  const float* v_max           = (const float*)d_in[4];
  const float* r_c             = (const float*)d_in[5];
  const float* Rd              = (const float*)d_in[6];
  const float* hz              = (const float*)d_in[7];
  const float* freqs           = (const float*)d_in[8];

  char* ws = (char*)d_ws;
  Params* prm  = (Params*)ws;                       // 256 B slot
  float*  vlab = (float*)(ws + 256);                // 128 f32
  float*  Rg   = (float*)(ws + 256 + 512);          // 1000 f32 (4096 B slot)
  float*  vg   = (float*)(ws + 256 + 512 + 4096);   // 1000 f32 (4096 B slot)
  float*  G    = (float*)(ws + 256 + 512 + 8192);   // 128*512 f32 = 256 KB

  setup_kernel<<<1, 256, 0, stream>>>(inclination, sky_rot, line_broadening,
                                      velocity_shift, v_max, r_c, Rd, hz, freqs,
                                      prm, vlab, Rg, vg);
  g_kernel<<<F_HI, B_BINS, 0, stream>>>(prm, vlab, G);
  cube_kernel<<<dim3(N_HI / 4, N_HI / 4), 256, 0, stream>>>(prm, Rg, vg, G,
                                                            (float*)d_out);
}